// SkipGramModel_32487132627200
// MI455X (gfx1250) — compile-verified
//
#include <hip/hip_runtime.h>
#include <math.h>
#include <stdint.h>

typedef __attribute__((ext_vector_type(2))) float v2f;
typedef __attribute__((ext_vector_type(4))) float v4f;
typedef __attribute__((ext_vector_type(8))) float v8f;

#define VOCAB  200000
#define DIM    128
#define NPOS   10
#define NNEG   64
#define KCHUNK 128           // vocab columns staged per chunk (VOCAB%4==0, klen in {128,64})
#define LDSPAD 4             // row stride 132 floats = 528B: 16B-aligned rows, bank-spread

// ---- CDNA5 async global->LDS copy (ASYNCcnt-tracked, 16B per lane) ----------
__device__ __forceinline__ void async_copy16(unsigned lds_off, const float* gaddr) {
    asm volatile("global_load_async_to_lds_b128 %0, %1, off"
                 :: "v"(lds_off), "v"(gaddr) : "memory");
}
__device__ __forceinline__ void wait_async() {
    asm volatile("s_wait_asynccnt 0x0" ::: "memory");
}
__device__ __forceinline__ unsigned lds_offset(const void* p) {
    return (unsigned)(uintptr_t)p;       // generic->LDS aperture: low 32 bits
}

// ---------------------------------------------------------------------------
// Kernel A: C1[16,128] += Wc[16,V] x IE^T[V,128]
//   row 0 = center_word, rows 1..10 = positive_words, rows 11..15 = zero pad
//   IE is [128, V] row-major, so B[k,n] = IE[n*V + k]
// ---------------------------------------------------------------------------
__global__ __launch_bounds__(256) void gemm_ctx(const float* __restrict__ center,
                                                const float* __restrict__ pos,
                                                const float* __restrict__ IE,
                                                float* __restrict__ C1, int nChunks) {
    __shared__ float sA[16][KCHUNK + LDSPAD];
    const int tid  = threadIdx.x;
    const int lane = tid & 31;
    const int wave = tid >> 5;            // 8 waves -> 8 N-tiles of 16 dims
    const int half = lane >> 4;
    const int idx  = lane & 15;           // M for A-frag, N for B-frag
    const int koff = half * 2;            // lane-half selects K {0,1} vs {2,3}
    const int nbase = wave * 16;

    // pad rows 11..15 are constant zero: write once, never touched again
    for (int i = tid; i < 5 * KCHUNK; i += 256)
        sA[11 + (i >> 7)][i & (KCHUNK - 1)] = 0.f;

    v8f acc = {};
    for (int c = blockIdx.x; c < nChunks; c += gridDim.x) {
        const int k0   = c * KCHUNK;
        const int klen = min(KCHUNK, VOCAB - k0);   // 128 or 64 (multiple of 4)
        __syncthreads();
        // DMA rows 0..10 into LDS: 11 * 32 = 352 16B blocks
        for (int b = tid; b < 11 * (KCHUNK / 4); b += 256) {
            const int r  = b >> 5;                  // 32 blocks per row
            const int cc = (b & 31) * 4;
            if (cc < klen) {
                const float* g = (r == 0) ? (center + k0 + cc)
                                          : (pos + (size_t)(r - 1) * VOCAB + k0 + cc);
                async_copy16(lds_offset(&sA[r][cc]), g);
            } else {
                *(v4f*)&sA[r][cc] = (v4f){0.f, 0.f, 0.f, 0.f};   // tail zero-fill
            }
        }
        wait_async();
        __syncthreads();

        const float* brow = IE + (size_t)(nbase + idx) * VOCAB + k0;
        if (c + gridDim.x < nChunks)                // global_prefetch_b8 next chunk
            __builtin_prefetch(brow + (size_t)gridDim.x * KCHUNK, 0, 1);

#pragma unroll 4
        for (int kk = 0; kk < klen; kk += 4) {
            v2f a = *(const v2f*)&sA[idx][kk + koff];     // A: W[m, k..k+1]
            v2f b = *(const v2f*)(brow + kk + koff);      // B: IE[n, k..k+1]
            acc = __builtin_amdgcn_wmma_f32_16x16x4_f32(
                false, a, false, b, (short)0, acc, false, false);
        }
    }
    // D layout: VGPR r, lanes 0-15 -> M=r, lanes 16-31 -> M=8+r; N = lane&15
    const int mbase = half * 8;
#pragma unroll
    for (int r = 0; r < 8; ++r)                       // rows 11..15 add exact zeros
        atomicAdd(&C1[(mbase + r) * DIM + nbase + idx], acc[r]);
}

// ---------------------------------------------------------------------------
// Kernel B: C2[64,128] += N[64,V] x OE[V,128]
//   OE is [V, 128] row-major, so B[k,n] = OE[k*128 + n] (n-contiguous, coalesced)
// ---------------------------------------------------------------------------
__global__ __launch_bounds__(256) void gemm_neg(const float* __restrict__ neg,
                                                const float* __restrict__ OE,
                                                float* __restrict__ C2, int nChunks) {
    __shared__ float sA[64][KCHUNK + LDSPAD];
    const int tid  = threadIdx.x;
    const int lane = tid & 31;
    const int wave = tid >> 5;
    const int half = lane >> 4;
    const int idx  = lane & 15;
    const int koff = half * 2;
    const int nbase = wave * 16;

    v8f acc[4] = {{}, {}, {}, {}};          // 4 M-tiles of 16 rows
    for (int c = blockIdx.x; c < nChunks; c += gridDim.x) {
        const int k0   = c * KCHUNK;
        const int klen = min(KCHUNK, VOCAB - k0);
        __syncthreads();
        // DMA all 64 rows: 64 * 32 = 2048 16B blocks (8 per thread)
        for (int b = tid; b < 64 * (KCHUNK / 4); b += 256) {
            const int r  = b >> 5;
            const int cc = (b & 31) * 4;
            if (cc < klen)
                async_copy16(lds_offset(&sA[r][cc]), neg + (size_t)r * VOCAB + k0 + cc);
            else
                *(v4f*)&sA[r][cc] = (v4f){0.f, 0.f, 0.f, 0.f};
        }
        wait_async();
        __syncthreads();

        if (c + gridDim.x < nChunks)        // prefetch next OE block start
            __builtin_prefetch(OE + (size_t)(k0 + (size_t)gridDim.x * KCHUNK) * DIM
                                  + nbase + idx, 0, 1);

#pragma unroll 2
        for (int kk = 0; kk < klen; kk += 4) {
            v2f b;
            b.x = OE[(size_t)(k0 + kk + koff)     * DIM + nbase + idx];
            b.y = OE[(size_t)(k0 + kk + koff + 1) * DIM + nbase + idx];
#pragma unroll
            for (int mt = 0; mt < 4; ++mt) {
                v2f a = *(const v2f*)&sA[mt * 16 + idx][kk + koff];
                acc[mt] = __builtin_amdgcn_wmma_f32_16x16x4_f32(
                    false, a, false, b, (short)0, acc[mt], false, false);
            }
        }
    }
    const int mbase = half * 8;
#pragma unroll
    for (int mt = 0; mt < 4; ++mt)
#pragma unroll
        for (int r = 0; r < 8; ++r)
            atomicAdd(&C2[(mt * 16 + mbase + r) * DIM + nbase + idx], acc[mt][r]);
}

// ---------------------------------------------------------------------------
// Finisher: 74 dot products of length 128 + stable log-sigmoid reduction
// ---------------------------------------------------------------------------
__device__ __forceinline__ float log_sigmoid(float x) {
    return (x < 0.f) ? (x - log1pf(expf(x))) : (-log1pf(expf(-x)));
}

__global__ __launch_bounds__(128) void finish(const float* __restrict__ C1,
                                              const float* __restrict__ C2,
                                              float* __restrict__ out) {
    __shared__ float sh[DIM];
    __shared__ float total;
    const int t = threadIdx.x;            // 128 threads = 4 waves
    if (t == 0) total = 0.f;
    sh[t] = C1[t];                        // h = center embedding (row 0)
    __syncthreads();

    const int lane = t & 31, wave = t >> 5;
    float lsum = 0.f;
    for (int s = wave; s < NPOS + NNEG; s += 4) {
        const float* row = (s < NPOS) ? (C1 + (size_t)(1 + s) * DIM)
                                      : (C2 + (size_t)(s - NPOS) * DIM);
        float p = 0.f;
        for (int d = lane; d < DIM; d += 32) p += sh[d] * row[d];
        for (int off = 16; off > 0; off >>= 1) p += __shfl_down(p, off, 32);
        if (lane == 0) {
            const float x = (s < NPOS) ? p : -p;
            lsum += log_sigmoid(x);
        }
    }
    if (lane == 0) atomicAdd(&total, lsum);
    __syncthreads();
    if (t == 0) *out = -total;
}

// ---------------------------------------------------------------------------
extern "C" void kernel_launch(void* const* d_in, const int* in_sizes, int n_in,
                              void* d_out, int out_size, void* d_ws, size_t ws_size,
                              hipStream_t stream) {
    const float* center = (const float*)d_in[0];   // [200000]
    const float* pos    = (const float*)d_in[1];   // [10, 200000]
    const float* neg    = (const float*)d_in[2];   // [64, 200000]
    const float* IE     = (const float*)d_in[3];   // [128, 200000]
    const float* OE     = (const float*)d_in[4];   // [200000, 128]

    float* C1 = (float*)d_ws;            // [16, 128] partial accumulators
    float* C2 = C1 + 16 * DIM;           // [64, 128]
    hipMemsetAsync(d_ws, 0, (16 + 64) * DIM * sizeof(float), stream);

    const int nChunks = (VOCAB + KCHUNK - 1) / KCHUNK;   // 1563
    gemm_ctx<<<512, 256, 0, stream>>>(center, pos, IE, C1, nChunks);
    gemm_neg<<<512, 256, 0, stream>>>(neg, OE, C2, nChunks);
    finish<<<1, 128, 0, stream>>>(C1, C2, (float*)d_out);
}